// DivEncLayer_4174708212236
// MI455X (gfx1250) — compile-verified
//
#include <hip/hip_runtime.h>

// CDNA5 / gfx1250: wave32, WMMA f32 16x16x4, async global->LDS staging,
// folded LayerNorm-affine + output projection, bias folded into WMMA C-init.
typedef __attribute__((ext_vector_type(2))) float v2f;
typedef __attribute__((ext_vector_type(4))) float v4f;
typedef __attribute__((ext_vector_type(8))) float v8f;
typedef int v4i_ __attribute__((__vector_size__(16)));   // builtin's expected pointee

#define QN 128
#define SN 8
#define UN 32
#define EPSV 1e-3f

#define WAVES 4
#define SLICES_PER_WAVE (QN / WAVES)        // 32
#define CHUNK 8                             // slices staged per async chunk
#define NCHUNK (SLICES_PER_WAVE / CHUNK)    // 4
#define ROWPAD 68                           // 64 floats per chunk-row + 4 pad (bank spread)
#define OUTPAD (QN + 1)                     // padded out tile row

#if __has_builtin(__builtin_amdgcn_global_load_async_to_lds_b128)
#define HAVE_ASYNC 1
#else
#define HAVE_ASYNC 0
#endif

#if HAVE_ASYNC
#if __has_builtin(__builtin_amdgcn_s_wait_asynccnt)
#define WAIT_ASYNC(n) do { __builtin_amdgcn_s_wait_asynccnt(n); asm volatile("" ::: "memory"); } while (0)
#else
#define WAIT_ASYNC(n) asm volatile("s_wait_asynccnt %0" :: "i"(n) : "memory")
#endif
typedef __attribute__((address_space(1))) v4i_* gptr16;
typedef __attribute__((address_space(3))) v4i_* lptr16;
#endif

struct F8 { float v[8]; };
__device__ __forceinline__ F8 ld8(const float* p) {
    F8 r;
    v4f a = *(const v4f*)p;
    v4f b = *(const v4f*)(p + 4);
    r.v[0]=a.x; r.v[1]=a.y; r.v[2]=a.z; r.v[3]=a.w;
    r.v[4]=b.x; r.v[5]=b.y; r.v[6]=b.z; r.v[7]=b.w;
    return r;
}

__global__ __launch_bounds__(WAVES * 32) void divenc_kernel(
    const float* __restrict__ x,      // (B, Q*S)
    const float* __restrict__ W1,     // (Q, S, U)
    const float* __restrict__ b1,     // (Q, U)
    const float* __restrict__ gamma,  // (Q, U)
    const float* __restrict__ beta,   // (Q, U)
    const float* __restrict__ W2,     // (Q, U)
    const float* __restrict__ b2,     // (Q,)
    float* __restrict__ out,          // (B, Q)
    int B)
{
#if HAVE_ASYNC
    __shared__ float xbuf[WAVES][2][16 * ROWPAD];   // double-buffered x columns per wave
#endif
    __shared__ float outTile[16 * OUTPAD];          // padded: writes hit distinct banks
    __shared__ float gwS[QN * UN];                  // gamma*W2, block-shared
    __shared__ float sgwS[QN];                      // sum_u gamma*W2 per slice
    __shared__ float dbS[QN];                       // sum_u beta*W2 + b2 per slice

    const int lane = threadIdx.x & 31;
    const int wave = threadIdx.x >> 5;
    const int half = lane >> 4;          // 0: lanes 0-15, 1: lanes 16-31
    const int l16  = lane & 15;
    const int rowBase = blockIdx.x * 16;
    const int qStart  = wave * SLICES_PER_WAVE;

#if HAVE_ASYNC
    // Stage one chunk: 16 rows x 64 floats (8 slices) into xbuf[wave][sel].
    // Instruction i covers rows {2i, 2i+1}: 32 lanes x 16B = 2 rows x 64 floats.
    auto stage = [&](int c, int sel) {
        const int colBase = (qStart + c * CHUNK) * SN;   // float offset in x row
        #pragma unroll
        for (int i = 0; i < 8; ++i) {
            const int row = 2 * i + half;
            int gr = rowBase + row;
            if (gr > B - 1) gr = B - 1;
            float* src = const_cast<float*>(x) + (size_t)gr * (QN * SN) + colBase + l16 * 4;
            float* dst = &xbuf[wave][sel][row * ROWPAD + l16 * 4];
            __builtin_amdgcn_global_load_async_to_lds_b128(
                (gptr16)src, (lptr16)dst, 0, 0);
        }
        asm volatile("" ::: "memory");
    };
    stage(0, 0);   // DMA in flight while we precompute folded coefficients
#endif

    // ---- Precompute gw = gamma*W2 (overlaps with first async chunk) ----
    for (int idx = threadIdx.x; idx < QN * UN; idx += blockDim.x)
        gwS[idx] = gamma[idx] * W2[idx];
    __syncthreads();
    // per-slice scalars: sgw = sum gw ; db = sum beta*W2 + b2
    for (int q = threadIdx.x; q < QN; q += blockDim.x) {
        float sg = 0.f, db = 0.f;
        #pragma unroll 4
        for (int u = 0; u < UN; ++u) {
            sg += gwS[q * UN + u];
            db += beta[q * UN + u] * W2[q * UN + u];
        }
        sgwS[q] = sg;
        dbS[q]  = db + b2[q];
    }
    __syncthreads();

    for (int c = 0; c < NCHUNK; ++c) {
#if HAVE_ASYNC
        if (c + 1 < NCHUNK) {
            stage(c + 1, (c + 1) & 1);
            WAIT_ASYNC(8);      // 8 outstanding = next chunk; this chunk is complete
        } else {
            WAIT_ASYNC(0);
        }
        const float* __restrict__ xb = &xbuf[wave][c & 1][0];
#endif
        for (int qc = 0; qc < CHUNK; ++qc) {
            const int q = qStart + c * CHUNK + qc;
            const float* __restrict__ w1q = W1 + q * (SN * UN);
            if (qc == 0) __builtin_prefetch(w1q + SN * UN, 0, 0);

            // This lane owns batch row rowBase+l16 with units:
            //   c0[m] -> u = half*8 + m ;  c1[m] -> u = 16 + half*8 + m
            const int u0 = q * UN + half * 8;
            const int u1 = q * UN + 16 + half * 8;
            const F8 B0  = ld8(b1 + u0),  B1  = ld8(b1 + u1);    // global, cached
            const F8 GW0 = ld8(gwS + u0), GW1 = ld8(gwS + u1);   // LDS broadcast
            const float sgwq = sgwS[q];
            const float dbq  = dbS[q];

            // ---- A fragments = W1[q]^T (16x4 f32 tiles; M=unit, K=s) ----
            // lanes 0-15: K = kk*4+{0,1}; lanes 16-31: K = kk*4+{2,3}; M = nt*16 + l16
            v2f a00, a01, a10, a11;   // a[kk][nt]
            {
                const int r0 = (half * 2) * UN;        // kk = 0
                a00.x = w1q[r0 + l16];        a00.y = w1q[r0 + UN + l16];
                a01.x = w1q[r0 + 16 + l16];   a01.y = w1q[r0 + UN + 16 + l16];
                const int r1 = (4 + half * 2) * UN;    // kk = 1
                a10.x = w1q[r1 + l16];        a10.y = w1q[r1 + UN + l16];
                a11.x = w1q[r1 + 16 + l16];  a11.y = w1q[r1 + UN + 16 + l16];
            }
            // ---- B fragments = x^T (4x16 f32; K=s, N=batch row) ----
            v2f bk0, bk1;
#if HAVE_ASYNC
            {
                const float* bp = xb + l16 * ROWPAD + qc * SN + half * 2;
                bk0.x = bp[0]; bk0.y = bp[1];     // K = half*2 + {0,1}
                bk1.x = bp[4]; bk1.y = bp[5];     // K = 4 + half*2 + {0,1}
            }
#else
            {
                int gr = rowBase + l16;
                if (gr > B - 1) gr = B - 1;
                const float* bp = x + (size_t)gr * (QN * SN) + q * SN + half * 2;
                bk0.x = bp[0]; bk0.y = bp[1];
                bk1.x = bp[4]; bk1.y = bp[5];
            }
#endif
            // D = h^T = W1^T x^T + bias: bias folded into the C-init (uniform
            // across lanes within a half; N does not index the bias).
            v8f c0, c1;
            #pragma unroll
            for (int m = 0; m < 8; ++m) { c0[m] = B0.v[m]; c1[m] = B1.v[m]; }
            c0 = __builtin_amdgcn_wmma_f32_16x16x4_f32(false, a00, false, bk0, (short)0, c0, false, false);
            c0 = __builtin_amdgcn_wmma_f32_16x16x4_f32(false, a10, false, bk1, (short)0, c0, false, false);
            c1 = __builtin_amdgcn_wmma_f32_16x16x4_f32(false, a01, false, bk0, (short)0, c1, false, false);
            c1 = __builtin_amdgcn_wmma_f32_16x16x4_f32(false, a11, false, bk1, (short)0, c1, false, false);

            float h0[8], h1[8];
            float s = 0.f, ss = 0.f;
            #pragma unroll
            for (int m = 0; m < 8; ++m) {
                float a = c0[m];
                float b = c1[m];
                a = a > 0.f ? a : (__expf(a) - 1.f);   // ELU(alpha=1)
                b = b > 0.f ? b : (__expf(b) - 1.f);
                h0[m] = a; h1[m] = b;
                s  += a + b;
                ss += a * a + b * b;
            }
            // other 16 units of the same row live in lane^16
            s  += __shfl_xor(s, 16, 32);
            ss += __shfl_xor(ss, 16, 32);
            const float mu = s * (1.f / 32.f);
            float var = ss * (1.f / 32.f) - mu * mu;
            var = var < 0.f ? 0.f : var;
            const float rstd = rsqrtf(var + EPSV);

            // out = rstd*(sum h*gw - mu*sgw) + (sum beta*w2 + b2)
            float t = 0.f;
            #pragma unroll
            for (int m = 0; m < 8; ++m)
                t += h0[m] * GW0.v[m] + h1[m] * GW1.v[m];
            t += __shfl_xor(t, 16, 32);

            if (half == 0)
                outTile[l16 * OUTPAD + q] = rstd * (t - mu * sgwq) + dbq;
        }
    }

    __syncthreads();

    // Coalesced store of the 16x128 tile (contiguous in (B,Q) layout).
    const size_t base = (size_t)rowBase * QN;
    for (int idx = threadIdx.x; idx < 16 * QN; idx += blockDim.x) {
        const int r = idx >> 7;
        const int qq = idx & (QN - 1);
        if (rowBase + r < B) out[base + idx] = outTile[r * OUTPAD + qq];
    }
}

extern "C" void kernel_launch(void* const* d_in, const int* in_sizes, int n_in,
                              void* d_out, int out_size, void* d_ws, size_t ws_size,
                              hipStream_t stream) {
    const float* x     = (const float*)d_in[0];
    const float* W1    = (const float*)d_in[1];
    const float* b1    = (const float*)d_in[2];
    const float* gamma = (const float*)d_in[3];
    const float* beta  = (const float*)d_in[4];
    const float* W2    = (const float*)d_in[5];
    const float* b2    = (const float*)d_in[6];
    float* out = (float*)d_out;

    const int B = in_sizes[0] / (QN * SN);
    const int nblocks = (B + 15) / 16;
    divenc_kernel<<<nblocks, WAVES * 32, 0, stream>>>(x, W1, b1, gamma, beta, W2, b2, out, B);
}